// GraphPF_1503238553909
// MI455X (gfx1250) — compile-verified
//
#include <hip/hip_runtime.h>

typedef __attribute__((ext_vector_type(2))) float v2f;
typedef __attribute__((ext_vector_type(8))) float v8f;

#define GB 16
#define GQ 64
#define GN 16384
#define GD 128
#define FMAXV 3.402823466e38f
#define K4CHUNK 2048
#define LSTR (GD + 4)   // padded LDS row stride: lane l hits bank (l*4+kb)%64 -> conflict-free

// Order-preserving float <-> uint mapping for atomicMax on signed floats.
__device__ __forceinline__ unsigned f2key(float f) {
    unsigned u = __float_as_uint(f);
    return (u & 0x80000000u) ? ~u : (u | 0x80000000u);
}
__device__ __forceinline__ float key2f(unsigned k) {
    unsigned u = (k & 0x80000000u) ? (k & 0x7FFFFFFFu) : ~k;
    return __uint_as_float(u);
}

// Low 32 bits of a generic pointer to __shared__ == LDS byte address
// (LDS aperture occupies addr[63:32]; addr[31:0] is the LDS offset).
__device__ __forceinline__ unsigned lds_addr(const void* p) {
    return (unsigned)(unsigned long long)p;
}

// CDNA5 async copy: 16 bytes global -> LDS, tracked by ASYNCcnt.
__device__ __forceinline__ void async_copy_b128(unsigned lds_byte, const float* gptr) {
    asm volatile("global_load_async_to_lds_b128 %0, %1, off"
                 :: "v"(lds_byte), "v"(gptr) : "memory");
}
__device__ __forceinline__ void wait_async0() {
    asm volatile("s_wait_asynccnt 0x0" ::: "memory");
}

// ---------------------------------------------------------------- K0: init
__global__ __launch_bounds__(256) void k0_init(unsigned* __restrict__ rowMaxKey,
                                               float* __restrict__ rowSum,
                                               float* __restrict__ out_res) {
    int i = blockIdx.x * blockDim.x + threadIdx.x;   // grid sized to GB*GQ*GD
    if (i < GB * GQ) {
        rowMaxKey[i] = f2key(-FMAXV);
        rowSum[i] = 0.0f;
    }
    out_res[i] = 0.0f;
}

// ------------------------------------------- K1: logits = Q*A^T via WMMA f32
// Block: 256 thr = 8 waves over a 64q x 32n tile. Q tile (64x128) and A tile
// (32x128) staged in padded LDS via global_load_async_to_lds_b128, then each
// wave runs 32 x v_wmma_f32_16x16x4_f32 fed by ds_load_b64.
__global__ __launch_bounds__(256) void k1_logits_wmma(
    const float* __restrict__ Qv, const float* __restrict__ Av,
    const int* __restrict__ node_nums,
    float* __restrict__ out_logits, unsigned* __restrict__ rowMaxKey) {
    __shared__ float sQ[GQ * LSTR];   // 64 rows, 33.8 KB
    __shared__ float sA[32 * LSTR];   // 32 rows, 16.9 KB

    const int b    = blockIdx.y;
    const int tid  = threadIdx.x;
    const int lane = tid & 31;
    const int wave = tid >> 5;
    const int nblk = blockIdx.x * 32;

    const float* gQ = Qv + (long)b * GQ * GD;
    const float* gA = Av + ((long)b * GN + nblk) * GD;

    // Stage Q tile: 64 rows x 32 float4-quads = 2048 quads, 8 per thread.
#pragma unroll
    for (int i = 0; i < 8; ++i) {
        const int idx = tid + i * 256;
        const int row = idx >> 5, qd = idx & 31;
        async_copy_b128(lds_addr(&sQ[row * LSTR + qd * 4]), gQ + row * GD + qd * 4);
    }
    // Stage A tile: 32 rows x 32 quads = 1024 quads, 4 per thread.
#pragma unroll
    for (int i = 0; i < 4; ++i) {
        const int idx = tid + i * 256;
        const int row = idx >> 5, qd = idx & 31;
        async_copy_b128(lds_addr(&sA[row * LSTR + qd * 4]), gA + row * GD + qd * 4);
    }
    wait_async0();
    __syncthreads();

    const int q0 = (wave & 3) * 16;     // q-strip of this wave
    const int nw = (wave >> 2) * 16;    // node-strip of this wave
    const int ml = lane & 15;           // M for A-op, N for B-op
    const int kb = (lane >> 4) * 2;     // K pair selected by lane half

    const float* sqrow = &sQ[(q0 + ml) * LSTR];
    const float* sarow = &sA[(nw + ml) * LSTR];

    v8f c = {0.f, 0.f, 0.f, 0.f, 0.f, 0.f, 0.f, 0.f};
#pragma unroll 8
    for (int k = 0; k < GD; k += 4) {
        v2f a  = *(const v2f*)(sqrow + k + kb);   // (M=ml, K=k+kb .. +1)
        v2f bb = *(const v2f*)(sarow + k + kb);   // (N=ml, K=k+kb .. +1)
        c = __builtin_amdgcn_wmma_f32_16x16x4_f32(false, a, false, bb,
                                                  (short)0, c, false, false);
    }

    const int nn   = node_nums[b];
    const int mh   = (lane >> 4) * 8;   // C/D layout: VGPR r -> M = r + 8*(lane>>4)
    const int ncol = nblk + nw + ml;
#pragma unroll
    for (int r = 0; r < 8; ++r) {
        const int m = mh + r;
        out_logits[((long)b * GQ + q0 + m) * GN + ncol] = c[r];
        float mv = (ncol < nn) ? c[r] : -FMAXV;
        // max across the 16 lanes of this half (xor<16 stays inside the half)
        for (int off = 8; off; off >>= 1)
            mv = fmaxf(mv, __shfl_xor(mv, off, 32));
        if (ml == 0)
            atomicMax(rowMaxKey + b * GQ + q0 + m, f2key(mv));
    }
}

// ---------------------------------------------------------- K2: finalize max
__global__ __launch_bounds__(256) void k2_maxfin(const unsigned* __restrict__ rowMaxKey,
                                                 float* __restrict__ rowMax) {
    int i = blockIdx.x * blockDim.x + threadIdx.x;
    if (i < GB * GQ) rowMax[i] = key2f(rowMaxKey[i]);
}

// ----------------------------------- K3: e = exp(masked - max), partial sums
__global__ __launch_bounds__(256) void k3_expsum(
    const float* __restrict__ logits, const int* __restrict__ node_nums,
    const float* __restrict__ rowMax, float* __restrict__ soft_un,
    float* __restrict__ rowSum) {
    __shared__ float red[256];
    const int b = blockIdx.z, q = blockIdx.y;
    const int base = blockIdx.x * 4096;
    const long row = ((long)b * GQ + q) * GN;
    const float rm = rowMax[b * GQ + q];
    const int nn = node_nums[b];
    float s = 0.0f;
    for (int i = threadIdx.x; i < 4096; i += 256) {
        const int n = base + i;
        const float x = logits[row + n];
        const float e = (n < nn) ? __expf(x - rm) : 0.0f;
        soft_un[row + n] = e;
        s += e;
    }
    red[threadIdx.x] = s;
    __syncthreads();
    for (int off = 128; off; off >>= 1) {
        if (threadIdx.x < off) red[threadIdx.x] += red[threadIdx.x + off];
        __syncthreads();
    }
    if (threadIdx.x == 0) atomicAdd(rowSum + b * GQ + q, red[0]);
}

// ----------------------------------------- K4: res = (e * A) / rowSum (WMMA)
// Block: per (batch, 2048-node K-chunk). Waves: (w&3)->q-strip, (w>>2)->64 d cols.
__global__ __launch_bounds__(256) void k4_gemm2_wmma(
    const float* __restrict__ Av, const float* __restrict__ soft_un,
    const float* __restrict__ rowSum, float* __restrict__ out_res) {
    const int b    = blockIdx.y;
    const int lane = threadIdx.x & 31;
    const int wave = threadIdx.x >> 5;
    const int q0   = (wave & 3) * 16;
    const int d0   = (wave >> 2) * 64;
    const int ml   = lane & 15;
    const int kb   = (lane >> 4) * 2;

    const float* prow  = soft_un + ((long)b * GQ + q0 + ml) * GN + (long)blockIdx.x * K4CHUNK;
    const float* abase = Av + ((long)b * GN + (long)blockIdx.x * K4CHUNK) * GD;

    v8f a0 = {0.f,0.f,0.f,0.f,0.f,0.f,0.f,0.f};
    v8f a1 = a0, a2 = a0, a3 = a0;

#pragma unroll 2
    for (int k = 0; k < K4CHUNK; k += 4) {
        if ((k & 63) == 0 && (k + 64) < K4CHUNK)
            __builtin_prefetch((const void*)(abase + (long)(k + 64 + kb) * GD + d0 + ml), 0, 1);
        v2f a = *(const v2f*)(prow + k + kb);            // A-op: (M=ml, K=k+kb..+1)
        const float* ak = abase + (long)(k + kb) * GD;   // rows K=k+kb and K=k+kb+1
        v2f b0, b1, b2, b3;                              // B-op: (N = d, K pair)
        b0.x = ak[d0 + 0  + ml]; b0.y = ak[GD + d0 + 0  + ml];
        b1.x = ak[d0 + 16 + ml]; b1.y = ak[GD + d0 + 16 + ml];
        b2.x = ak[d0 + 32 + ml]; b2.y = ak[GD + d0 + 32 + ml];
        b3.x = ak[d0 + 48 + ml]; b3.y = ak[GD + d0 + 48 + ml];
        a0 = __builtin_amdgcn_wmma_f32_16x16x4_f32(false, a, false, b0, (short)0, a0, false, false);
        a1 = __builtin_amdgcn_wmma_f32_16x16x4_f32(false, a, false, b1, (short)0, a1, false, false);
        a2 = __builtin_amdgcn_wmma_f32_16x16x4_f32(false, a, false, b2, (short)0, a2, false, false);
        a3 = __builtin_amdgcn_wmma_f32_16x16x4_f32(false, a, false, b3, (short)0, a3, false, false);
    }

    const int mh = (lane >> 4) * 8;
#pragma unroll
    for (int r = 0; r < 8; ++r) {
        const int row = b * GQ + q0 + mh + r;
        const float inv = 1.0f / rowSum[row];
        float* orow = out_res + (long)row * GD + d0 + ml;
        atomicAdd(orow + 0,  a0[r] * inv);
        atomicAdd(orow + 16, a1[r] * inv);
        atomicAdd(orow + 32, a2[r] * inv);
        atomicAdd(orow + 48, a3[r] * inv);
    }
}

// ------------------------------------------------ K5: normalize prob_soft
__global__ __launch_bounds__(256) void k5_norm(float* __restrict__ soft,
                                               const float* __restrict__ rowSum) {
    const int b = blockIdx.z, q = blockIdx.y;
    const float inv = 1.0f / rowSum[b * GQ + q];
    const long idx = ((long)b * GQ + q) * GN + (long)blockIdx.x * 1024 + threadIdx.x * 4;
    float4 v = *(const float4*)(soft + idx);
    v.x *= inv; v.y *= inv; v.z *= inv; v.w *= inv;
    *(float4*)(soft + idx) = v;
}

extern "C" void kernel_launch(void* const* d_in, const int* in_sizes, int n_in,
                              void* d_out, int out_size, void* d_ws, size_t ws_size,
                              hipStream_t stream) {
    const float* Qv = (const float*)d_in[0];     // [B,Q,D]
    const float* Av = (const float*)d_in[1];     // [B,N,D]
    const int*   nn = (const int*)d_in[2];       // [B]

    float* out        = (float*)d_out;
    float* out_res    = out;                                   // [B,Q,D]
    float* out_soft   = out + (long)GB * GQ * GD;              // [B,Q,N]
    float* out_logits = out_soft + (long)GB * GQ * GN;         // [B,Q,N]

    unsigned* rowMaxKey = (unsigned*)d_ws;                     // [B*Q]
    float*    rowMax    = (float*)d_ws + 1024;                 // [B*Q]
    float*    rowSum    = (float*)d_ws + 2048;                 // [B*Q]

    k0_init<<<dim3((GB * GQ * GD) / 256), 256, 0, stream>>>(rowMaxKey, rowSum, out_res);
    k1_logits_wmma<<<dim3(GN / 32, GB), 256, 0, stream>>>(Qv, Av, nn, out_logits, rowMaxKey);
    k2_maxfin<<<dim3(4), 256, 0, stream>>>(rowMaxKey, rowMax);
    k3_expsum<<<dim3(GN / 4096, GQ, GB), 256, 0, stream>>>(out_logits, nn, rowMax, out_soft, rowSum);
    k4_gemm2_wmma<<<dim3(GN / K4CHUNK, GB), 256, 0, stream>>>(Av, out_soft, rowSum, out_res);
    k5_norm<<<dim3(GN / 1024, GQ, GB), 256, 0, stream>>>(out_soft, rowSum);
}